// CurricularFace_86655260164559
// MI455X (gfx1250) — compile-verified
//
#include <hip/hip_runtime.h>
#include <hip/hip_bf16.h>
#include <stdint.h>

// ---- CurricularFace constants (margin = 0.5, s = 64) ----
#define COS_M_F    0.8775825618903728f
#define SIN_M_F    0.4794255386042030f
#define THRESH_F  -0.8775825618903728f   // cos(pi - m) = -cos(m)
#define MM_F       0.2397127693021015f   // sin(pi - m) * m
#define S_F        64.0f

typedef __attribute__((ext_vector_type(2)))  float        v2f;
typedef __attribute__((ext_vector_type(8)))  float        v8f;
typedef __attribute__((ext_vector_type(4)))  unsigned int v4u;
typedef __attribute__((ext_vector_type(4)))  int          v4i;
typedef __attribute__((ext_vector_type(8)))  int          v8i;

#if defined(__has_builtin) && __has_builtin(__builtin_amdgcn_s_wait_tensorcnt)
#define WAIT_TENSORCNT(n) __builtin_amdgcn_s_wait_tensorcnt(n)
#else
#define WAIT_TENSORCNT(n) asm volatile("s_wait_tensorcnt %0" :: "i"(n) : "memory")
#endif

__device__ __forceinline__ float cf_clip(float v) {
    return fminf(fmaxf(v, -1.0f), 1.0f);
}

__device__ __forceinline__ float cf_transform(float v, int col, float ctm, float ftl,
                                              bool valid, int lab, float tn) {
    float x = cf_clip(v);
    float y = (valid && (x > ctm)) ? x * (tn + x) : x;
    if (valid && (col == lab)) y = ftl;
    return y * S_F;
}

// ---------------------------------------------------------------------------
// Kernel 1: per-row target-logit stats + n_valid (single block, 512 threads)
// ---------------------------------------------------------------------------
__global__ void cf_rowstats(const float* __restrict__ logits,
                            const int* __restrict__ labels,
                            float* __restrict__ ctmA, float* __restrict__ ftlA,
                            float* __restrict__ valA, int* __restrict__ labA,
                            float* __restrict__ nvalidOut, int B, int C) {
    __shared__ float red[512];
    int tid = threadIdx.x;
    float cnt = 0.0f;
    for (int b = tid; b < B; b += (int)blockDim.x) {
        int lab    = labels[b];
        bool valid = (lab >= 0);
        int  l2    = valid ? lab : 0;
        float x    = cf_clip(logits[(size_t)b * (size_t)C + (size_t)l2]);
        float sn   = sqrtf(fmaxf(0.0f, 1.0f - x * x));
        float ctm  = x * COS_M_F - sn * SIN_M_F;
        float ftl  = (x > THRESH_F) ? ctm : (x - MM_F);
        ctmA[b] = ctm;
        ftlA[b] = ftl;
        valA[b] = valid ? 1.0f : 0.0f;
        labA[b] = lab;
        cnt += valid ? 1.0f : 0.0f;
    }
    red[tid] = cnt;
    __syncthreads();
    for (int s = 256; s > 0; s >>= 1) {
        if (tid < s) red[tid] += red[tid + s];
        __syncthreads();
    }
    if (tid == 0) nvalidOut[0] = red[0];
}

// ---------------------------------------------------------------------------
// Kernel 2: partial sums of clip(logits) over valid rows.
// V_WMMA_F32_16X16X4_F32 with B = ones folds 64 elements per issue:
// D[m][n] = sum_k A[m][k]. Each lane does ONE b128 load per 128-element
// wave-group and feeds two wmma ops (fast path, wave-uniform branch so EXEC
// is all-1s at the wmma). Lane-sum of the 8 accumulators + shfl_xor(16)
// (wave32 halves) recovers the wave total in every lane.
// ---------------------------------------------------------------------------
__global__ void cf_reduce(const float* __restrict__ logits,
                          const float* __restrict__ valA,
                          float* __restrict__ partials,
                          int C, int chunksPerRow) {
    int row       = blockIdx.y;
    int chunkSize = (C + chunksPerRow - 1) / chunksPerRow;  // multiple of 4 for C=100000
    int start     = (int)blockIdx.x * chunkSize;
    int len       = min(chunkSize, C - start);
    const float* base = logits + (size_t)row * (size_t)C + (size_t)start;

    int tid  = threadIdx.x;
    int wave = tid >> 5;
    int lane = tid & 31;

    v8f acc = {0.f, 0.f, 0.f, 0.f, 0.f, 0.f, 0.f, 0.f};
    v2f ones; ones.x = 1.0f; ones.y = 1.0f;

    for (int g = wave; g * 128 < len; g += 8) {   // wave-uniform loop
        int e = g * 128 + lane * 4;
        v2f a0, a1;
        if (g * 128 + 128 <= len) {               // wave-uniform: full group, b128 load
            float4 v = *(const float4*)(base + e);
            a0.x = cf_clip(v.x); a0.y = cf_clip(v.y);
            a1.x = cf_clip(v.z); a1.y = cf_clip(v.w);
        } else {                                  // tail group: guarded scalar loads
            int e0 = min(e,     len - 1), e1 = min(e + 1, len - 1);
            int e2 = min(e + 2, len - 1), e3 = min(e + 3, len - 1);
            float x0 = cf_clip(base[e0]); if (e     >= len) x0 = 0.0f;
            float x1 = cf_clip(base[e1]); if (e + 1 >= len) x1 = 0.0f;
            float x2 = cf_clip(base[e2]); if (e + 2 >= len) x2 = 0.0f;
            float x3 = cf_clip(base[e3]); if (e + 3 >= len) x3 = 0.0f;
            a0.x = x0; a0.y = x1; a1.x = x2; a1.y = x3;
        }
        acc = __builtin_amdgcn_wmma_f32_16x16x4_f32(
            false, a0, false, ones, (short)0, acc, false, false);
        acc = __builtin_amdgcn_wmma_f32_16x16x4_f32(
            false, a1, false, ones, (short)0, acc, false, false);
    }

    float s = acc[0] + acc[1] + acc[2] + acc[3] + acc[4] + acc[5] + acc[6] + acc[7];
    s += __shfl_xor(s, 16, 32);                   // join the two 16-lane halves

    __shared__ float wsum[8];
    if (lane == 0) wsum[wave] = s;
    __syncthreads();
    if (tid == 0) {
        float t = 0.0f;
        #pragma unroll
        for (int w = 0; w < 8; ++w) t += wsum[w];
        partials[(size_t)blockIdx.y * gridDim.x + blockIdx.x] = t * valA[row];
    }
}

// ---------------------------------------------------------------------------
// Kernel 3: finalize t_new = mean_valid * 0.01 + 0.99 * t[0]
// ---------------------------------------------------------------------------
__global__ void cf_finalize(const float* __restrict__ partials, int nPartials,
                            const float* __restrict__ nvalidPtr,
                            const float* __restrict__ t_in,
                            float* __restrict__ tnewOut, int C) {
    __shared__ float red[256];
    int tid = threadIdx.x;
    float s = 0.0f;
    for (int i = tid; i < nPartials; i += 256) s += partials[i];
    red[tid] = s;
    __syncthreads();
    for (int k = 128; k > 0; k >>= 1) {
        if (tid < k) red[tid] += red[tid + k];
        __syncthreads();
    }
    if (tid == 0) {
        float nv   = nvalidPtr[0];
        float mean = red[0] / (nv * (float)C);
        tnewOut[0] = mean * 0.01f + 0.99f * t_in[0];
    }
}

// ---------------------------------------------------------------------------
// Kernel 4: main streaming pass, TDM double-buffered through LDS.
// Each block owns a quarter-row; wave 0 issues tensor_load_to_lds descriptors
// (1-row tile, data_size=4B) and synchronizes via s_wait_tensorcnt + barriers.
// TDM builtin called UNCONDITIONALLY this round (probe signal on purpose).
// ---------------------------------------------------------------------------
#define TILE_FLOATS 4096   // 16 KB per buffer; 2 buffers = 32 KB -> 10 blocks/WGP

__device__ __forceinline__ void cf_tdm_issue(const float* gsrc, const float* lds_dst,
                                             int tw) {
    uint64_t ga  = (uint64_t)(uintptr_t)gsrc;
    uint32_t lo  = (uint32_t)(uintptr_t)lds_dst;   // low 32 bits = LDS offset
    unsigned utw = (unsigned)tw;
    // D# group 0: count=1 | lds_addr | global_addr[56:0] | type=2
    v4u g0 = { 1u, lo, (uint32_t)ga,
               ((uint32_t)(ga >> 32) & 0x1FFFFFFu) | (2u << 30) };
    // D# group 1: data_size=4B; tensor_dim0=tw, tensor_dim1=1;
    //             tile_dim0=tw (1-row tile); tensor_dim0_stride=tw
    v8i g1 = { (int)(2u << 16),
               (int)((utw & 0xFFFFu) << 16),
               (int)(((utw >> 16) & 0xFFFFu) | (1u << 16)),
               (int)((utw & 0xFFFFu) << 16),
               0,
               (int)utw,
               0, 0 };
    v4i g2 = {0, 0, 0, 0};
    v4i g3 = {0, 0, 0, 0};
#if defined(__clang_major__) && (__clang_major__ >= 23)
    v8i g4 = {0, 0, 0, 0, 0, 0, 0, 0};
    __builtin_amdgcn_tensor_load_to_lds(g0, g1, g2, g3, g4, 0);
#else
    __builtin_amdgcn_tensor_load_to_lds(g0, g1, g2, g3, 0);
#endif
}

__global__ void cf_main(const float* __restrict__ logits,
                        const float* __restrict__ ctmA,
                        const float* __restrict__ ftlA,
                        const float* __restrict__ valA,
                        const int* __restrict__ labA,
                        const float* __restrict__ tnewPtr,
                        float* __restrict__ out,
                        int C, int qlen) {
    __shared__ __align__(16) float buf[2][TILE_FLOATS];

    int row     = blockIdx.y;
    int colBase = (int)blockIdx.x * qlen;
    int nfloats = min(qlen, C - colBase);
    if (nfloats <= 0) return;
    int ntiles  = (nfloats + TILE_FLOATS - 1) / TILE_FLOATS;
    int tid     = threadIdx.x;

    float ctm   = ctmA[row];
    float ftl   = ftlA[row];
    bool  valid = (valA[row] > 0.5f);
    int   lab   = labA[row];
    float tn    = tnewPtr[0];

    const float* rowSrc = logits + (size_t)row * (size_t)C + (size_t)colBase;
    float*       rowDst = out    + (size_t)row * (size_t)C + (size_t)colBase;

    if (tid < 32)
        cf_tdm_issue(rowSrc, &buf[0][0], min(TILE_FLOATS, nfloats));

    for (int t = 0; t < ntiles; ++t) {
        int tstart = t * TILE_FLOATS;
        int tw     = min(TILE_FLOATS, nfloats - tstart);

        if (tid < 32) {
            if (t + 1 < ntiles) {
                int ns = (t + 1) * TILE_FLOATS;
                cf_tdm_issue(rowSrc + ns, &buf[(t + 1) & 1][0],
                             min(TILE_FLOATS, nfloats - ns));
                WAIT_TENSORCNT(1);      // tile t done (TDM ops in-order per wave)
            } else {
                WAIT_TENSORCNT(0);
            }
        }
        __syncthreads();                // tile t visible to all waves

        const float* lsrc = &buf[t & 1][0];
        int nf4 = tw >> 2;
        for (int j = tid; j < nf4; j += (int)blockDim.x) {
            float4 v = ((const float4*)lsrc)[j];
            int c0 = colBase + tstart + 4 * j;
            float4 r;
            r.x = cf_transform(v.x, c0 + 0, ctm, ftl, valid, lab, tn);
            r.y = cf_transform(v.y, c0 + 1, ctm, ftl, valid, lab, tn);
            r.z = cf_transform(v.z, c0 + 2, ctm, ftl, valid, lab, tn);
            r.w = cf_transform(v.w, c0 + 3, ctm, ftl, valid, lab, tn);
            ((float4*)(rowDst + tstart))[j] = r;
        }
        int rem = tw & 3;               // scalar tail (tw not multiple of 4)
        if (tid < rem) {
            int e = (nf4 << 2) + tid;
            rowDst[tstart + e] =
                cf_transform(lsrc[e], colBase + tstart + e, ctm, ftl, valid, lab, tn);
        }
        __syncthreads();                // all reads of buf[t&1] done before reuse
    }
}

// ---------------------------------------------------------------------------
extern "C" void kernel_launch(void* const* d_in, const int* in_sizes, int n_in,
                              void* d_out, int out_size, void* d_ws, size_t ws_size,
                              hipStream_t stream) {
    const float* logits = (const float*)d_in[0];
    const int*   labels = (const int*)d_in[1];
    const float* t      = (const float*)d_in[2];
    float*       outp   = (float*)d_out;

    int B = in_sizes[1];
    int C = in_sizes[0] / B;

    const int chunksPerRow = 8;

    // workspace layout (floats): [0]=n_valid, [1]=t_new, [4..)=partials,
    // then per-row ctm / ftl / valid / lab arrays.
    float* wsf      = (float*)d_ws;
    float* nvalid   = wsf + 0;
    float* tnew     = wsf + 1;
    float* partials = wsf + 4;
    float* ctmA     = partials + (size_t)chunksPerRow * B;
    float* ftlA     = ctmA + B;
    float* valA     = ftlA + B;
    int*   labA     = (int*)(valA + B);

    cf_rowstats<<<1, 512, 0, stream>>>(logits, labels, ctmA, ftlA, valA, labA,
                                       nvalid, B, C);

    dim3 g2(chunksPerRow, B);
    cf_reduce<<<g2, 256, 0, stream>>>(logits, valA, partials, C, chunksPerRow);

    cf_finalize<<<1, 256, 0, stream>>>(partials, chunksPerRow * B, nvalid, t,
                                       tnew, C);

    const int NQ = 4;                                  // row quarters
    int qlen = (((C + NQ - 1) / NQ) + 3) & ~3;         // multiple of 4 floats
    dim3 g4((C + qlen - 1) / qlen, B);
    cf_main<<<g4, 256, 0, stream>>>(logits, ctmA, ftlA, valA, labA, tnew,
                                    outp, C, qlen);
}